// RandomProjectionQuantizer_88390426952410
// MI455X (gfx1250) — compile-verified
//
#include <hip/hip_runtime.h>
#include <hip/hip_bf16.h>

// ---------------------------------------------------------------------------
// RandomProjectionQuantizer for MI455X (gfx1250, wave32)
//   1. compact mask -> ordered indices (deterministic prefix sum, 1 WG)
//   2. c2[k] = ||code_k||^2
//   3. targets = gathered_rows @ W^T      (memory-bound, VALU + b128 loads)
//   4. t2[n] = ||target_n||^2
//   5. fused distance GEMM + argmin via V_WMMA_F32_16X16X4_F32, two 16-code
//      tiles per iteration, software-pipelined B loads, per-row running
//      minima (index tie-break deferred to final merge). No (N,K) distance
//      matrix ever touches memory.
//   6. finalize: write argmin index as float
// ---------------------------------------------------------------------------

typedef __attribute__((ext_vector_type(2))) float v2f;
typedef __attribute__((ext_vector_type(8))) float v8f;

#define NMASK     16384
#define BL        32768
#define DIM       320
#define CODE_DIM  16
#define NUM_CODES 8192
#define ROW_TILES (NMASK / 16)      // 1024 waves, one 16-row tile each
#define CODE_TILES (NUM_CODES / 16) // 512 code tiles total

// ---------------------------------------------------------------- init
__global__ void rpq_init(unsigned long long* packed) {
    *packed = 0xFFFFFFFFFFFFFFFFull;
}

// ---------------------------------------------------------------- compact
// One workgroup of 1024 threads; thread t owns mask[t*32 .. t*32+31].
// Serial exclusive scan of the 1024 per-thread counts keeps the gather
// order identical to jnp.nonzero (increasing flat index) -> argmin
// tie-breaking matches the reference.
__global__ __launch_bounds__(1024)
void rpq_compact(const int* __restrict__ mask, int* __restrict__ idx) {
    __shared__ int cnt[1024];
    const int t = threadIdx.x;
    int c = 0;
    #pragma unroll
    for (int e = 0; e < 32; ++e) c += (mask[t * 32 + e] != 0);
    cnt[t] = c;
    __syncthreads();
    if (t == 0) {
        int run = 0;
        for (int i = 0; i < 1024; ++i) { int v = cnt[i]; cnt[i] = run; run += v; }
    }
    __syncthreads();
    int pos = cnt[t];
    for (int e = 0; e < 32; ++e) {
        int flat = t * 32 + e;
        if (mask[flat] != 0) idx[pos++] = flat;
    }
}

// ---------------------------------------------------------------- c2
__global__ void rpq_c2(const float* __restrict__ cb, float* __restrict__ c2) {
    int k = blockIdx.x * blockDim.x + threadIdx.x;
    const float* p = cb + k * CODE_DIM;
    float s = 0.f;
    #pragma unroll
    for (int c = 0; c < CODE_DIM; ++c) s += p[c] * p[c];
    c2[k] = s;
}

// ---------------------------------------------------------------- projection
// gid -> (row n, out-dim c). 16 adjacent lanes share a row -> broadcast loads,
// the 21 MB of gathered rows stream from HBM exactly once (cache reuse).
__global__ void rpq_proj(const float* __restrict__ x, const int* __restrict__ idx,
                         const float* __restrict__ w, float* __restrict__ tg) {
    int gid = blockIdx.x * blockDim.x + threadIdx.x;
    int n = gid >> 4;
    int c = gid & 15;
    const float4* r  = (const float4*)(x + (long)idx[n] * DIM);
    const float4* wc = (const float4*)(w + c * DIM);
    float s = 0.f;
    #pragma unroll 8
    for (int d4 = 0; d4 < DIM / 4; ++d4) {
        float4 a = r[d4], b = wc[d4];
        s += a.x * b.x + a.y * b.y + a.z * b.z + a.w * b.w;
    }
    tg[n * CODE_DIM + c] = s;
}

// ---------------------------------------------------------------- t2
__global__ void rpq_t2(const float* __restrict__ tg, float* __restrict__ t2) {
    int n = blockIdx.x * blockDim.x + threadIdx.x;
    const float* p = tg + n * CODE_DIM;
    float s = 0.f;
    #pragma unroll
    for (int c = 0; c < CODE_DIM; ++c) s += p[c] * p[c];
    t2[n] = s;
}

// ---------------------------------------------------------------- fused GEMM+argmin
// Wave32: each wave owns one 16-row tile of targets; loops over the codebook
// 32 codes (two 16-code tiles) per iteration. Each 16x16x16 f32 GEMM is four
// chained V_WMMA_F32_16X16X4_F32 ops. B operands for the NEXT iteration are
// loaded while the current WMMAs + epilogue run (distinct registers -> the
// s_wait lands a full tile later). Running minima are kept per output row so
// the in-loop update is cmp+2*cndmask; strict '<' with k ascending preserves
// first-occurrence semantics within a row, and the final 8-way merge applies
// the full (value, index) lexicographic rule across rows.
//
// Layouts (ISA 7.12.2, 32-bit operands):
//   A 16x4 : lane = M (0..15 twice); VGPR0 K = khalf, VGPR1 K = khalf+1,
//            khalf = 2*(lane>=16)
//   B 4x16 : lane = N (0..15 twice); same K striping
//   C/D    : vgpr v, lane l -> (M = v + 8*(l>=16), N = l&15)
__global__ __launch_bounds__(256)
void rpq_dist_argmin(const float* __restrict__ tg, const float* __restrict__ t2,
                     const float* __restrict__ cb, const float* __restrict__ c2,
                     unsigned long long* __restrict__ packed) {
    const int lane   = threadIdx.x & 31;
    const int waveId = (blockIdx.x * blockDim.x + threadIdx.x) >> 5; // 0..1023
    const int hi     = lane >> 4;            // 0 / 1
    const int m      = lane & 15;            // A: row-in-tile, B/D: col
    const int khalf  = hi << 1;

    // --- A operands: 16x16 target tile, sliced into 4 K-chunks of 4
    const float* T = tg + waveId * (16 * CODE_DIM);
    v2f a[4];
    #pragma unroll
    for (int j = 0; j < 4; ++j) {
        a[j].x = T[m * CODE_DIM + 4 * j + khalf];
        a[j].y = T[m * CODE_DIM + 4 * j + khalf + 1];
    }
    // --- t2 for the rows this lane sees in the D layout
    float t2row[8];
    #pragma unroll
    for (int v = 0; v < 8; ++v)
        t2row[v] = t2[waveId * 16 + v + 8 * hi];

    float minv[8];
    int   mink[8];
    #pragma unroll
    for (int v = 0; v < 8; ++v) { minv[v] = 3.402823466e+38f; mink[v] = 0; }

    // --- double-buffered B operands for two 16-code tiles per iteration
    v2f   bA[4], bB[4];
    float c2A, c2B;
    {
        const float* BtA = cb + (0 + m) * CODE_DIM;
        const float* BtB = cb + (16 + m) * CODE_DIM;
        #pragma unroll
        for (int j = 0; j < 4; ++j) {
            bA[j].x = BtA[4 * j + khalf];  bA[j].y = BtA[4 * j + khalf + 1];
            bB[j].x = BtB[4 * j + khalf];  bB[j].y = BtB[4 * j + khalf + 1];
        }
        c2A = c2[0 + m];
        c2B = c2[16 + m];
    }

    const int ITER = CODE_TILES / 2;            // 256 iterations, 32 codes each
    for (int it = 0; it < ITER; ++it) {
        const int k0  = it * 32;
        const int nk0 = ((it + 1) & (ITER - 1)) * 32;  // wrap -> branchless tail
        __builtin_prefetch(cb + ((it + 2) & (ITER - 1)) * 32 * CODE_DIM, 0, 3);

        // issue next iteration's loads into fresh registers (overlap w/ WMMA)
        v2f   nA[4], nB[4];
        float nc2A, nc2B;
        {
            const float* BtA = cb + (nk0 + m) * CODE_DIM;
            const float* BtB = cb + (nk0 + 16 + m) * CODE_DIM;
            #pragma unroll
            for (int j = 0; j < 4; ++j) {
                nA[j].x = BtA[4 * j + khalf];  nA[j].y = BtA[4 * j + khalf + 1];
                nB[j].x = BtB[4 * j + khalf];  nB[j].y = BtB[4 * j + khalf + 1];
            }
            nc2A = c2[nk0 + m];
            nc2B = c2[nk0 + 16 + m];
        }

        // --- two 16x16x16 GEMM tiles, K chained over 4 WMMAs each
        v8f accA = {}, accB = {};
        #pragma unroll
        for (int j = 0; j < 4; ++j)
            accA = __builtin_amdgcn_wmma_f32_16x16x4_f32(
                false, a[j], false, bA[j], (short)0, accA, false, false);
        #pragma unroll
        for (int j = 0; j < 4; ++j)
            accB = __builtin_amdgcn_wmma_f32_16x16x4_f32(
                false, a[j], false, bB[j], (short)0, accB, false, false);

        // --- slim epilogue: per-row running min, no index math in loop
        const int kA = k0 + m;
        const int kB = k0 + 16 + m;
        #pragma unroll
        for (int v = 0; v < 8; ++v) {
            float dA = __builtin_amdgcn_fmed3f(
                fmaf(-2.0f, accA[v], t2row[v] + c2A), 0.0f, 3.402823466e+38f);
            if (dA < minv[v]) { minv[v] = dA; mink[v] = kA; }
            float dB = __builtin_amdgcn_fmed3f(
                fmaf(-2.0f, accB[v], t2row[v] + c2B), 0.0f, 3.402823466e+38f);
            if (dB < minv[v]) { minv[v] = dB; mink[v] = kB; }
        }

        #pragma unroll
        for (int j = 0; j < 4; ++j) { bA[j] = nA[j]; bB[j] = nB[j]; }
        c2A = nc2A;  c2B = nc2B;
    }

    // --- merge the 8 per-row minima with full (value, index) tie-breaking
    float    bestv = minv[0];
    unsigned besti = (unsigned)((waveId * 16 + 0 + 8 * hi) * NUM_CODES + mink[0]);
    #pragma unroll
    for (int v = 1; v < 8; ++v) {
        unsigned fi = (unsigned)((waveId * 16 + v + 8 * hi) * NUM_CODES + mink[v]);
        bool take = (minv[v] < bestv) || (minv[v] == bestv && fi < besti);
        bestv = take ? minv[v] : bestv;
        besti = take ? fi : besti;
    }

    // --- wave32 reduction (lexicographic on (value, index))
    #pragma unroll
    for (int off = 16; off > 0; off >>= 1) {
        float    ov = __shfl_xor(bestv, off, 32);
        unsigned oi = __shfl_xor(besti, off, 32);
        bool take = (ov < bestv) || (ov == bestv && oi < besti);
        bestv = take ? ov : bestv;
        besti = take ? oi : besti;
    }
    if (lane == 0) {
        // clamped d2 >= 0 -> IEEE bit pattern is order-preserving as unsigned
        unsigned long long key =
            ((unsigned long long)__float_as_uint(bestv) << 32) | (unsigned long long)besti;
        atomicMin(packed, key);
    }
}

// ---------------------------------------------------------------- finalize
__global__ void rpq_finalize(const unsigned long long* packed, float* out) {
    out[0] = (float)(unsigned)(*packed & 0xFFFFFFFFull);
}

// ---------------------------------------------------------------------------
extern "C" void kernel_launch(void* const* d_in, const int* in_sizes, int n_in,
                              void* d_out, int out_size, void* d_ws, size_t ws_size,
                              hipStream_t stream) {
    const float* input = (const float*)d_in[0];   // (16,2048,320) f32
    const int*   mask  = (const int*)  d_in[1];   // (16,2048)     i32
    const float* W     = (const float*)d_in[2];   // (16,320)      f32
    const float* CB    = (const float*)d_in[3];   // (8192,16)     f32
    float* out = (float*)d_out;

    char* ws = (char*)d_ws;
    unsigned long long* packed = (unsigned long long*)ws;
    int*   idx     = (int*)  (ws + 256);
    float* t2      = (float*)(ws + 256 + 65536);
    float* c2      = (float*)(ws + 256 + 65536 + 65536);
    float* targets = (float*)(ws + 256 + 65536 + 65536 + 32768);

    rpq_init<<<dim3(1), dim3(1), 0, stream>>>(packed);
    rpq_compact<<<dim3(1), dim3(1024), 0, stream>>>(mask, idx);
    rpq_c2<<<dim3(NUM_CODES / 256), dim3(256), 0, stream>>>(CB, c2);
    rpq_proj<<<dim3((NMASK * CODE_DIM) / 256), dim3(256), 0, stream>>>(input, idx, W, targets);
    rpq_t2<<<dim3(NMASK / 256), dim3(256), 0, stream>>>(targets, t2);
    rpq_dist_argmin<<<dim3(ROW_TILES / 8), dim3(256), 0, stream>>>(targets, t2, CB, c2, packed);
    rpq_finalize<<<dim3(1), dim3(1), 0, stream>>>(packed, out);
}